// KroneckerProjection_64587718197487
// MI455X (gfx1250) — compile-verified
//
#include <hip/hip_runtime.h>

typedef __attribute__((ext_vector_type(2))) float v2f;
typedef __attribute__((ext_vector_type(8))) float v8f;

#define PDIM 128
#define QDIM 16
#define RDIM 16
#define SDIM 512
#define DIN  2048
#define DOUT 8192
#define NROWS (4 * 4096)

#define WAVES_PER_BLOCK 8
#define BLOCK_THREADS 256

// LDS operand packing (wave32 WMMA 16x16x4 f32 layouts):
//  A-operand (16x4): VGPR v, lane L -> M = L%16, K = v + 2*(L/16)
//  B-operand (4x16): VGPR v, lane L -> N = L%16, K = v + 2*(L/16)
//  C/D (16x16):      VGPR v, lane L -> M = v + 8*(L/16), N = L%16
// Packed chunk = 64 floats, element index = lane*2 + v  -> ds_load_b64/lane,
// banks {2L, 2L+1} cover all 64 banks exactly once (conflict-free).

__global__ __launch_bounds__(BLOCK_THREADS)
void kron_proj_wmma_kernel(const float* __restrict__ x,
                           const float* __restrict__ A,
                           const float* __restrict__ B,
                           float* __restrict__ out)
{
    __shared__ float sAop[32 * 64];                 // 8 KB:  A^T operand chunks (i in blocks of 4)
    __shared__ float sBop[128 * 64];                // 32 KB: B operand chunks (lt,kc)
    __shared__ float sY[WAVES_PER_BLOCK][16 * 18];  // 9 KB:  per-wave Y transpose scratch

    const int tid    = threadIdx.x;
    const int wave   = tid >> 5;
    const int lane   = tid & 31;
    const int laneLo = lane & 15;
    const int laneHi = lane >> 4;

    // ---- Stage A (P x Q) as A^T WMMA A-operand chunks: chunk ic covers i = 4*ic..4*ic+3
    for (int t = tid; t < 32 * 64; t += BLOCK_THREADS) {
        int ic = t >> 6, r = t & 63;
        int L = r >> 1, v = r & 1;
        int i = 4 * ic + v + 2 * (L >> 4);
        int j = L & 15;
        sAop[t] = A[i * QDIM + j];                  // A^T[j][i]
    }
    // ---- Stage B (R x S) as B-operand chunks: chunk c = lt*4 + kc
    for (int t = tid; t < 128 * 64; t += BLOCK_THREADS) {
        int c = t >> 6, r = t & 63;
        int L = r >> 1, v = r & 1;
        int lt = c >> 2, kc = c & 3;
        int k = 4 * kc + v + 2 * (L >> 4);
        int l = lt * 16 + (L & 15);
        sBop[t] = B[k * SDIM + l];
    }
    __syncthreads();

    float* y = &sY[wave][0];

    for (long long row = (long long)blockIdx.x * WAVES_PER_BLOCK + wave;
         row < NROWS;
         row += (long long)gridDim.x * WAVES_PER_BLOCK) {
        const float* __restrict__ xrow = x + row * DIN;

        // ---- Step 1: Y(16x16) = A^T(16x128) @ x4(128x16), K chunked by 4 ----
        v8f acc = {};
        #pragma unroll 4
        for (int ic = 0; ic < 32; ++ic) {
            v2f a = *(const v2f*)(&sAop[ic * 64 + lane * 2]);   // ds_load_b64
            // x B-operand: value(L,v) = x[n][64*ic + 16*(v + 2*laneHi) + laneLo]
            int base = 64 * ic + 32 * laneHi + laneLo;
            v2f b;
            b.x = xrow[base];
            b.y = xrow[base + 16];
            acc = __builtin_amdgcn_wmma_f32_16x16x4_f32(
                false, a, false, b, (short)0, acc, false, false);
        }

        // ---- Re-stripe Y: C/D layout -> A-operand layout via per-wave LDS ----
        #pragma unroll
        for (int v = 0; v < 8; ++v) {
            int m = v + 8 * laneHi;                 // Y row j
            y[m * 18 + laneLo] = acc[v];            // Y[j][k], pitch 18 (pad)
        }
        v2f ya[4];
        #pragma unroll
        for (int kc = 0; kc < 4; ++kc) {
            // A-operand chunk kc: lane L, vgpr v -> Y[L%16][4*kc + v + 2*laneHi]
            ya[kc] = *(const v2f*)(&y[laneLo * 18 + 4 * kc + 2 * laneHi]);
        }

        // ---- Step 2: out_row(16x512) = Y(16x16) @ B(16x512) ----
        float* __restrict__ orow = out + row * DOUT;
        for (int lt = 0; lt < 32; ++lt) {
            v8f c = {};
            #pragma unroll
            for (int kc = 0; kc < 4; ++kc) {
                v2f b = *(const v2f*)(&sBop[(lt * 4 + kc) * 64 + lane * 2]);
                c = __builtin_amdgcn_wmma_f32_16x16x4_f32(
                    false, ya[kc], false, b, (short)0, c, false, false);
            }
            // D tile: lane L, vgpr v -> out[n][(v + 8*laneHi)*512 + lt*16 + laneLo]
            #pragma unroll
            for (int v = 0; v < 8; ++v) {
                orow[(v + 8 * laneHi) * SDIM + lt * 16 + laneLo] = c[v];
            }
        }
    }
}

extern "C" void kernel_launch(void* const* d_in, const int* in_sizes, int n_in,
                              void* d_out, int out_size, void* d_ws, size_t ws_size,
                              hipStream_t stream) {
    const float* x = (const float*)d_in[0];
    const float* A = (const float*)d_in[1];
    const float* B = (const float*)d_in[2];
    float* out = (float*)d_out;

    dim3 grid(NROWS / WAVES_PER_BLOCK);   // 2048 blocks, 8 waves each -> 1 row per wave
    dim3 block(BLOCK_THREADS);
    hipLaunchKernelGGL(kron_proj_wmma_kernel, grid, block, 0, stream, x, A, B, out);
}